// LDM_BI_70145405878282
// MI455X (gfx1250) — compile-verified
//
#include <hip/hip_runtime.h>

typedef __attribute__((ext_vector_type(2))) float v2f;
typedef __attribute__((ext_vector_type(8))) float v8f;

#define EPS 1e-6f
#define DLAT 8
#define LOG2E 1.44269504088896340736f

// ---------------------------------------------------------------------------
// Dense normalizer: sum_{i,j} exp(gamma_i + delta_j - ||(z_i+eps) - q_j|| - eps)
// One wave32 per 16x16 tile; K=8 done as two V_WMMA_F32_16X16X4_F32.
// ---------------------------------------------------------------------------
__global__ void __launch_bounds__(256) ldm_dense_kernel(
    const float* __restrict__ z, const float* __restrict__ q,
    const float* __restrict__ gamma, const float* __restrict__ delta,
    float* __restrict__ partials, int M)
{
    const int lane   = threadIdx.x & 31;
    const int wave   = threadIdx.x >> 5;
    const int nwaves = blockDim.x >> 5;
    const int i0     = blockIdx.x << 4;          // 16 rows per block

    const int  m  = lane & 15;                   // M for A, N for B/C
    const bool hi = (lane >= 16);                // K pair select: {0,1} vs {2,3}

    // A-matrix (z + eps), rows i0..i0+15, resident across the whole j sweep.
    const float4 zlo = *(const float4*)(z + (size_t)(i0 + m) * DLAT);
    const float4 zhi = *(const float4*)(z + (size_t)(i0 + m) * DLAT + 4);
    v2f a0, a1;
    a0.x = (hi ? zlo.z : zlo.x) + EPS;  a0.y = (hi ? zlo.w : zlo.y) + EPS;
    a1.x = (hi ? zhi.z : zhi.x) + EPS;  a1.y = (hi ? zhi.w : zhi.y) + EPS;

    // C/D layout: lanes 0-15 own rows i0..i0+7, lanes 16-31 own i0+8..i0+15;
    // column is j0 + m.
    const int rbase = i0 + (hi ? 8 : 0);
    float znv[8], gv[8];
#pragma unroll
    for (int v = 0; v < 8; ++v) {
        const float4 lo = *(const float4*)(z + (size_t)(rbase + v) * DLAT);
        const float4 hi4 = *(const float4*)(z + (size_t)(rbase + v) * DLAT + 4);
        float x0 = lo.x + EPS, x1 = lo.y + EPS, x2 = lo.z + EPS, x3 = lo.w + EPS;
        float x4 = hi4.x + EPS, x5 = hi4.y + EPS, x6 = hi4.z + EPS, x7 = hi4.w + EPS;
        znv[v] = x0*x0 + x1*x1 + x2*x2 + x3*x3 + x4*x4 + x5*x5 + x6*x6 + x7*x7;
        gv[v]  = gamma[rbase + v];
    }

    float lsum = 0.f;
    const int jtiles = M >> 4;
    for (int jt = wave; jt < jtiles; jt += nwaves) {
        const int col = (jt << 4) + m;
        const float4 qlo = *(const float4*)(q + (size_t)col * DLAT);
        const float4 qhi = *(const float4*)(q + (size_t)col * DLAT + 4);
        const float qn = qlo.x*qlo.x + qlo.y*qlo.y + qlo.z*qlo.z + qlo.w*qlo.w
                       + qhi.x*qhi.x + qhi.y*qhi.y + qhi.z*qhi.z + qhi.w*qhi.w;

        v2f b0, b1;                               // 4 cndmasks, no cmp chain
        b0.x = hi ? qlo.z : qlo.x;  b0.y = hi ? qlo.w : qlo.y;
        b1.x = hi ? qhi.z : qhi.x;  b1.y = hi ? qhi.w : qhi.y;

        v8f acc = {0.f, 0.f, 0.f, 0.f, 0.f, 0.f, 0.f, 0.f};
        acc = __builtin_amdgcn_wmma_f32_16x16x4_f32(false, a0, false, b0,
                                                    (short)0, acc, false, false);
        acc = __builtin_amdgcn_wmma_f32_16x16x4_f32(false, a1, false, b1,
                                                    (short)0, acc, false, false);

        const float base = delta[col] - EPS;
#pragma unroll
        for (int v = 0; v < 8; ++v) {
            float d2   = znv[v] + qn - 2.f * acc[v];
            float dist = __builtin_amdgcn_sqrtf(fmaxf(d2, 0.f));   // raw v_sqrt_f32
            lsum += __builtin_amdgcn_exp2f((gv[v] + base - dist) * LOG2E);
        }
    }

    // Deterministic in-block tree reduction.
    __shared__ float red[256];
    red[threadIdx.x] = lsum;
    __syncthreads();
    for (int s = 128; s > 0; s >>= 1) {
        if ((int)threadIdx.x < s) red[threadIdx.x] += red[threadIdx.x + s];
        __syncthreads();
    }
    if (threadIdx.x == 0) partials[blockIdx.x] = red[0];
}

// ---------------------------------------------------------------------------
// Sparse Poisson term: sum_e w_e * (gamma_i + delta_j - ||z_i - q_j + eps||)
// ---------------------------------------------------------------------------
__global__ void __launch_bounds__(256) ldm_sparse_kernel(
    const float* __restrict__ z, const float* __restrict__ q,
    const float* __restrict__ gamma, const float* __restrict__ delta,
    const float* __restrict__ w, const int* __restrict__ si,
    const int* __restrict__ sj, float* __restrict__ partials, int nnz)
{
    const int stride = gridDim.x * blockDim.x;
    float s = 0.f;
    for (int e = blockIdx.x * blockDim.x + threadIdx.x; e < nnz; e += stride) {
        const int i = si[e];
        const int j = sj[e];
        const float4 zlo = *(const float4*)(z + (size_t)i * DLAT);
        const float4 zhi = *(const float4*)(z + (size_t)i * DLAT + 4);
        const float4 qlo = *(const float4*)(q + (size_t)j * DLAT);
        const float4 qhi = *(const float4*)(q + (size_t)j * DLAT + 4);
        float d0 = zlo.x - qlo.x + EPS, d1 = zlo.y - qlo.y + EPS;
        float d2 = zlo.z - qlo.z + EPS, d3 = zlo.w - qlo.w + EPS;
        float d4 = zhi.x - qhi.x + EPS, d5 = zhi.y - qhi.y + EPS;
        float d6 = zhi.z - qhi.z + EPS, d7 = zhi.w - qhi.w + EPS;
        float ss = d0*d0 + d1*d1 + d2*d2 + d3*d3 + d4*d4 + d5*d5 + d6*d6 + d7*d7;
        s += w[e] * (gamma[i] + delta[j] - __builtin_amdgcn_sqrtf(ss));
    }
    __shared__ float red[256];
    red[threadIdx.x] = s;
    __syncthreads();
    for (int st = 128; st > 0; st >>= 1) {
        if ((int)threadIdx.x < st) red[threadIdx.x] += red[threadIdx.x + st];
        __syncthreads();
    }
    if (threadIdx.x == 0) partials[blockIdx.x] = red[0];
}

// ---------------------------------------------------------------------------
// Final deterministic combine: out = sum(sparse partials) - sum(dense partials)
// ---------------------------------------------------------------------------
__global__ void __launch_bounds__(256) ldm_reduce_final(
    const float* __restrict__ dpart, int nd,
    const float* __restrict__ spart, int ns, float* __restrict__ out)
{
    float s = 0.f;
    for (int t = threadIdx.x; t < ns; t += blockDim.x) s += spart[t];
    for (int t = threadIdx.x; t < nd; t += blockDim.x) s -= dpart[t];
    __shared__ float red[256];
    red[threadIdx.x] = s;
    __syncthreads();
    for (int st = 128; st > 0; st >>= 1) {
        if ((int)threadIdx.x < st) red[threadIdx.x] += red[threadIdx.x + st];
        __syncthreads();
    }
    if (threadIdx.x == 0) out[0] = red[0];
}

extern "C" void kernel_launch(void* const* d_in, const int* in_sizes, int n_in,
                              void* d_out, int out_size, void* d_ws, size_t ws_size,
                              hipStream_t stream) {
    const float* z     = (const float*)d_in[0];   // latent_z  [N, 8]
    const float* q     = (const float*)d_in[1];   // latent_q  [M, 8]
    const float* gamma = (const float*)d_in[2];   // [N]
    const float* delta = (const float*)d_in[3];   // [M]
    const float* w     = (const float*)d_in[4];   // [nnz]
    const int*   si    = (const int*)d_in[5];     // [nnz]
    const int*   sj    = (const int*)d_in[6];     // [nnz]

    const int N   = in_sizes[2];                  // 16384
    const int M   = in_sizes[3];                  // 8192
    const int nnz = in_sizes[4];                  // 2,000,000

    const int denseBlocks  = N >> 4;              // one 16-row tile per block
    const int sparseBlocks = 1024;

    float* dpart = (float*)d_ws;
    float* spart = dpart + denseBlocks;

    ldm_dense_kernel<<<denseBlocks, 256, 0, stream>>>(z, q, gamma, delta, dpart, M);
    ldm_sparse_kernel<<<sparseBlocks, 256, 0, stream>>>(z, q, gamma, delta, w, si, sj,
                                                        spart, nnz);
    ldm_reduce_final<<<1, 256, 0, stream>>>(dpart, denseBlocks, spart, sparseBlocks,
                                            (float*)d_out);
}